// lay_c_cbed_nni_out_43387759624559
// MI455X (gfx1250) — compile-verified
//
#include <hip/hip_runtime.h>
#include <hip/hip_bf16.h>

// Problem constants (from reference: x[32,512,512,4] f32, mk[32,512,512,1] f32)
#define NB 32
#define NH 512
#define NW 512
#define BLOCKS_PER_BATCH 64          // 512*512/64 = 4096 px per block
#define RED_THREADS 256              // -> 16 px per thread

typedef __attribute__((ext_vector_type(2))) float v2f;
typedef __attribute__((ext_vector_type(8))) float v8f;

// ---------------------------------------------------------------------------
// Cross-lane (wave32) sum via V_WMMA_F32_16X16X4_F32 with an all-ones B.
// A layout (ISA 7.12.2, 32-bit A 16x4): VGPR0 holds K=0 (lanes 0-15, M=lane)
// and K=2 (lanes 16-31, M=lane-16); VGPR1 holds K=1/K=3. With A = {val, 0}
// and B = ones, D[m][n] = val[m] + val[m+16]. A lane's 8 D VGPRs cover rows
// 0-7 (lanes<16) or 8-15 (lanes>=16), so sum8 + shfl_xor(16) = full wave sum.
// ---------------------------------------------------------------------------
__device__ __forceinline__ float wave_sum_wmma(float val) {
    v2f a;  a[0] = val;  a[1] = 0.0f;     // A: per-lane partial in K=0/K=2 slots
    v2f bo; bo[0] = 1.0f; bo[1] = 1.0f;   // B: all ones (layout-independent)
    v8f c = {};
    c = __builtin_amdgcn_wmma_f32_16x16x4_f32(
            /*neg_a=*/false, a, /*neg_b=*/false, bo,
            /*c_mod=*/(short)0, c, /*reuse_a=*/false, /*reuse_b=*/false);
    float s = ((c[0] + c[1]) + (c[2] + c[3])) + ((c[4] + c[5]) + (c[6] + c[7]));
    s += __shfl_xor(s, 16, 32);           // combine the two half-wave sums
    return s;
}

// ---------------------------------------------------------------------------
// Stage 1: per-block partial moments of mk.  grid = NB*BLOCKS_PER_BATCH
// Each thread accumulates 16 pixels (fixed order), LDS fold 256->32, then the
// first wave finishes the reduction through the WMMA tensor pipe.
// partial[blockIdx.x*3 + {0,1,2}] = {sum(mk), sum(rx*mk), sum(ry*mk)}
// ---------------------------------------------------------------------------
__global__ void moments_partial_kernel(const float* __restrict__ mk,
                                       float* __restrict__ partial) {
    const int b   = blockIdx.x / BLOCKS_PER_BATCH;
    const int blk = blockIdx.x % BLOCKS_PER_BATCH;
    const int tid = threadIdx.x;

    float s0 = 0.f, s1 = 0.f, s2 = 0.f;
    const int base = b * (NH * NW);
#pragma unroll
    for (int k = 0; k < 16; ++k) {
        const int idx = blk * 4096 + k * RED_THREADS + tid;   // 0..262143
        const float m = mk[base + idx];
        const int h = idx >> 9;          // NW == 512
        const int w = idx & (NW - 1);
        s0 += m;
        s1 += (float)(w - (NW / 2)) * m; // rx * mk
        s2 += (float)(h - (NH / 2)) * m; // ry * mk
    }

    __shared__ float l0[RED_THREADS], l1[RED_THREADS], l2[RED_THREADS];
    l0[tid] = s0; l1[tid] = s1; l2[tid] = s2;
    __syncthreads();

    if (tid < 32) {                       // wave 0, EXEC all ones for WMMA
        float a0 = 0.f, a1 = 0.f, a2 = 0.f;
#pragma unroll
        for (int j = 0; j < 8; ++j) {     // fixed order -> deterministic
            a0 += l0[tid + j * 32];
            a1 += l1[tid + j * 32];
            a2 += l2[tid + j * 32];
        }
        const float t0 = wave_sum_wmma(a0);
        const float t1 = wave_sum_wmma(a1);
        const float t2 = wave_sum_wmma(a2);
        if (tid == 0) {
            float* p = partial + (size_t)blockIdx.x * 3;
            p[0] = t0; p[1] = t1; p[2] = t2;
        }
    }
}

// ---------------------------------------------------------------------------
// Stage 2: fold the 64 per-block partials of each batch (fixed-order tree)
// and emit the integer shift (dx, dy).  grid = NB, block = 64.
// ---------------------------------------------------------------------------
__global__ void moments_final_kernel(const float* __restrict__ partial,
                                     int* __restrict__ shift) {
    const int b = blockIdx.x;
    const int t = threadIdx.x;            // 0..63
    __shared__ float l0[64], l1[64], l2[64];
    const float* p = partial + ((size_t)b * BLOCKS_PER_BATCH + t) * 3;
    l0[t] = p[0]; l1[t] = p[1]; l2[t] = p[2];
    __syncthreads();
    for (int off = 32; off > 0; off >>= 1) {
        if (t < off) { l0[t] += l0[t + off]; l1[t] += l1[t + off]; l2[t] += l2[t + off]; }
        __syncthreads();
    }
    if (t == 0) {
        const float m00 = fmaxf(0.001f, l0[0]);
        shift[b * 2 + 0] = (int)rintf(l1[0] / m00);   // dx (round half-to-even)
        shift[b * 2 + 1] = (int)rintf(l2[0] / m00);   // dy
    }
}

// ---------------------------------------------------------------------------
// Stage 3: nearest-neighbor shifted copy with zero-fill (pad+clip+gather of
// the reference collapses to this).  One float4 (one pixel, C=4) per thread:
// fully coalesced b128 traffic, ~256 MiB total -> HBM-bound (~12 us).
// grid = (NW/256, NH, NB), block = 256.
// ---------------------------------------------------------------------------
__global__ void nn_shift_kernel(const float4* __restrict__ x,
                                const int* __restrict__ shift,
                                float4* __restrict__ out) {
    const int b = blockIdx.z;
    const int h = blockIdx.y;
    const int w = blockIdx.x * blockDim.x + threadIdx.x;
    const int dx = shift[b * 2 + 0];      // uniform per batch -> scalar cached
    const int dy = shift[b * 2 + 1];
    const int sy = h + dy;
    const int sx = w + dx;
    float4 v = make_float4(0.f, 0.f, 0.f, 0.f);
    if ((unsigned)sy < (unsigned)NH && (unsigned)sx < (unsigned)NW)
        v = x[((size_t)b * NH + sy) * NW + sx];
    out[((size_t)b * NH + h) * NW + w] = v;
}

// ---------------------------------------------------------------------------
extern "C" void kernel_launch(void* const* d_in, const int* in_sizes, int n_in,
                              void* d_out, int out_size, void* d_ws, size_t ws_size,
                              hipStream_t stream) {
    const float* x  = (const float*)d_in[0];   // [32,512,512,4] f32
    const float* mk = (const float*)d_in[1];   // [32,512,512,1] f32
    float* out = (float*)d_out;                // [32,512,512,4] f32

    // Workspace layout: partial sums then per-batch integer shifts (~24.8 KB).
    float* partial = (float*)d_ws;                                   // NB*64*3 f32
    int*   shift   = (int*)((char*)d_ws +
                            (size_t)NB * BLOCKS_PER_BATCH * 3 * sizeof(float));

    moments_partial_kernel<<<NB * BLOCKS_PER_BATCH, RED_THREADS, 0, stream>>>(mk, partial);
    moments_final_kernel<<<NB, 64, 0, stream>>>(partial, shift);

    dim3 grid(NW / 256, NH, NB);
    nn_shift_kernel<<<grid, 256, 0, stream>>>((const float4*)x, shift, (float4*)out);
}